// HeteroGNN_76570676953326
// MI455X (gfx1250) — compile-verified
//
#include <hip/hip_runtime.h>

typedef __attribute__((ext_vector_type(2))) float v2f;
typedef __attribute__((ext_vector_type(8))) float v8f;

#define H 128
#define NS 50000
#define NP 100000
#define NW 20000
#define ES 500000
#define EST 500000
#define ESH 400000

static inline int cdiv(long long a, long long b) { return (int)((a + b - 1) / b); }

// ---------------- utility kernels ----------------

__global__ void zero_f32(float* __restrict__ p, long long n) {
    long long i = (long long)blockIdx.x * blockDim.x + threadIdx.x;
    if (i < n) p[i] = 0.0f;
}

// One wave (32 lanes) per edge; each lane atomically adds 4 consecutive floats.
__global__ void scatter_add(const float* __restrict__ hsrc,
                            const int* __restrict__ ei, int E,
                            float* __restrict__ acc, float* __restrict__ cnt) {
    long long t = (long long)blockIdx.x * blockDim.x + threadIdx.x;
    int e    = (int)(t >> 5);
    int lane = (int)(t & 31);
    if (e >= E) return;
    int src = ei[e];
    int dst = ei[E + e];
    const float* s = hsrc + (size_t)src * H + lane * 4;
    float*       a = acc  + (size_t)dst * H + lane * 4;
#pragma unroll
    for (int c = 0; c < 4; ++c) atomicAdd(&a[c], s[c]);
    if (lane == 0) atomicAdd(&cnt[dst], 1.0f);
}

// ---------------- WMMA f32 GEMM (projections): C = relu(A[M,K] @ W[K,128] + b) ----------
// One wave per 16-row strip, all 8 16x16 column tiles in registers.
__global__ void gemm_wmma_f32(const float* __restrict__ A,
                              const float* __restrict__ W,
                              const float* __restrict__ bias,
                              float* __restrict__ C,
                              int M, int K) {
    int wave = threadIdx.x >> 5;
    int lane = threadIdx.x & 31;
    int tile = blockIdx.x * (blockDim.x >> 5) + wave;
    int ntiles = M >> 4;
    if (tile >= ntiles) return;          // wave-uniform: EXEC stays all-1 for WMMA

    int m0    = tile << 4;
    int lm    = lane & 15;               // row (A) / col (B,C) within tile
    int khalf = (lane >> 4) << 1;        // 0 or 2: which K-pair this lane holds
    int mrow  = m0 + lm;                 // A row for this lane
    int mw    = m0 + ((lane >> 4) << 3); // C/D row base for this lane half

    v8f acc[8] = {};
    const float* arow = A + (size_t)mrow * K;
    for (int k = 0; k < K; k += 4) {
        v2f a;
        a.x = arow[k + khalf];
        a.y = arow[k + khalf + 1];
        const float* w0 = W + (size_t)(k + khalf) * H;
#pragma unroll
        for (int nt = 0; nt < 8; ++nt) {
            int ncol = nt * 16 + lm;
            v2f b;
            b.x = w0[ncol];
            b.y = w0[H + ncol];
            acc[nt] = __builtin_amdgcn_wmma_f32_16x16x4_f32(
                false, a, false, b, (short)0, acc[nt], false, false);
        }
    }

#pragma unroll
    for (int nt = 0; nt < 8; ++nt) {
        int ncol = nt * 16 + lm;
        float bv = bias[ncol];
#pragma unroll
        for (int r = 0; r < 8; ++r) {
            float v = fmaxf(acc[nt][r] + bv, 0.0f);
            C[(size_t)(mw + r) * H + ncol] = v;
        }
    }
}

// ---- Fused SAGE output GEMM:  C = relu( sum_t scale_t(A_t) @ W_t  + bias0 + bias1 )
// Terms 0,1 may carry a count vector (mean aggregation: rows scaled by 1/max(cnt,1)).
// K fixed at 128 for all terms. Single C write; no intermediate traffic.
__global__ void gemm_fused(const float* __restrict__ A0, const float* __restrict__ W0,
                           const float* __restrict__ c0,
                           const float* __restrict__ A1, const float* __restrict__ W1,
                           const float* __restrict__ c1,
                           const float* __restrict__ A2, const float* __restrict__ W2,
                           const float* __restrict__ A3, const float* __restrict__ W3,
                           const float* __restrict__ bias0, const float* __restrict__ bias1,
                           float* __restrict__ C, int M, int nterms) {
    int wave = threadIdx.x >> 5;
    int lane = threadIdx.x & 31;
    int tile = blockIdx.x * (blockDim.x >> 5) + wave;
    int ntiles = M >> 4;
    if (tile >= ntiles) return;

    int m0    = tile << 4;
    int lm    = lane & 15;
    int khalf = (lane >> 4) << 1;
    int mrow  = m0 + lm;
    int mw    = m0 + ((lane >> 4) << 3);

    const float* As[4] = {A0, A1, A2, A3};
    const float* Ws[4] = {W0, W1, W2, W3};
    const float* cs[4] = {c0, c1, nullptr, nullptr};

    v8f acc[8] = {};
    for (int t = 0; t < nterms; ++t) {
        const float* arow = As[t] + (size_t)mrow * H;
        float s = 1.0f;
        if (cs[t]) s = 1.0f / fmaxf(cs[t][mrow], 1.0f);   // fused mean
        for (int k = 0; k < H; k += 4) {
            v2f a;
            a.x = arow[k + khalf] * s;
            a.y = arow[k + khalf + 1] * s;
            const float* w0 = Ws[t] + (size_t)(k + khalf) * H;
#pragma unroll
            for (int nt = 0; nt < 8; ++nt) {
                int ncol = nt * 16 + lm;
                v2f b;
                b.x = w0[ncol];
                b.y = w0[H + ncol];
                acc[nt] = __builtin_amdgcn_wmma_f32_16x16x4_f32(
                    false, a, false, b, (short)0, acc[nt], false, false);
            }
        }
    }

#pragma unroll
    for (int nt = 0; nt < 8; ++nt) {
        int ncol = nt * 16 + lm;
        float bv = bias0[ncol] + (bias1 ? bias1[ncol] : 0.0f);
#pragma unroll
        for (int r = 0; r < 8; ++r) {
            float v = fmaxf(acc[nt][r] + bv, 0.0f);
            C[(size_t)(mw + r) * H + ncol] = v;
        }
    }
}

// ---------------- launch helpers ----------------

static void launch_proj(const float* A, const float* W, const float* b, float* C,
                        int M, int K, hipStream_t s) {
    gemm_wmma_f32<<<cdiv(M / 16, 8), 256, 0, s>>>(A, W, b, C, M, K);
}

extern "C" void kernel_launch(void* const* d_in, const int* in_sizes, int n_in,
                              void* d_out, int out_size, void* d_ws, size_t ws_size,
                              hipStream_t stream) {
    const float* x_sup  = (const float*)d_in[0];   // [NS,64]
    const float* x_prod = (const float*)d_in[1];   // [NP,32]
    const float* x_wh   = (const float*)d_in[2];   // [NW,16]
    const int* ei_sup   = (const int*)d_in[3];     // [2,ES]  supplier->product
    const int* ei_sto   = (const int*)d_in[4];     // [2,EST] product->warehouse
    const int* ei_shp   = (const int*)d_in[5];     // [2,ESH] warehouse->product

    // jax pytree flatten order: params{"layers":[...], "proj":{...}} ->
    // layers first (each layer dict sorted: ships, stored, supplies; each et: Wl, Wr, bl),
    // then proj sorted: product, supplier, warehouse (each: W, b).
    struct ET { const float *Wl, *Wr, *bl; };
    ET L[2][3];  // [layer][0=ships,1=stored,2=supplies]
    int p = 6;
    for (int l = 0; l < 2; ++l)
        for (int e = 0; e < 3; ++e) {
            L[l][e].Wl = (const float*)d_in[p++];
            L[l][e].Wr = (const float*)d_in[p++];
            L[l][e].bl = (const float*)d_in[p++];
        }
    const float* Wp_prod = (const float*)d_in[p++];
    const float* bp_prod = (const float*)d_in[p++];
    const float* Wp_sup  = (const float*)d_in[p++];
    const float* bp_sup  = (const float*)d_in[p++];
    const float* Wp_wh   = (const float*)d_in[p++];
    const float* bp_wh   = (const float*)d_in[p++];

    // output regions
    float* out_sup  = (float*)d_out;
    float* out_prod = out_sup + (size_t)NS * H;
    float* out_wh   = out_prod + (size_t)NP * H;

    // workspace layout
    float* w = (float*)d_ws;
    size_t o = 0;
    float* hp_a    = w + o; o += (size_t)NP * H;
    float* hw_a    = w + o; o += (size_t)NW * H;
    float* hp_b    = w + o; o += (size_t)NP * H;
    float* hw_b    = w + o; o += (size_t)NW * H;
    float* acc_sup = w + o; o += (size_t)NP * H;   // zeroed region starts here
    float* acc_shp = w + o; o += (size_t)NP * H;
    float* acc_sto = w + o; o += (size_t)NW * H;
    float* cnt_sup = w + o; o += NP;
    float* cnt_shp = w + o; o += NP;
    float* cnt_sto = w + o; o += NW;
    long long zlen = (long long)(2 * (size_t)NP * H + (size_t)NW * H + 2 * NP + NW);

    // ---- projections (ReLU fused). h_sup is final (relu idempotent, never updated).
    launch_proj(x_sup,  Wp_sup,  bp_sup,  out_sup, NS, 64, stream);
    launch_proj(x_prod, Wp_prod, bp_prod, hp_a,    NP, 32, stream);
    launch_proj(x_wh,   Wp_wh,   bp_wh,   hw_a,    NW, 16, stream);

    float* hp = hp_a;
    float* hw = hw_a;
    for (int layer = 0; layer < 2; ++layer) {
        float* hp_new = (layer == 0) ? hp_b : out_prod;
        float* hw_new = (layer == 0) ? hw_b : out_wh;

        zero_f32<<<cdiv(zlen, 256), 256, 0, stream>>>(acc_sup, zlen);

        scatter_add<<<cdiv((long long)ES  * 32, 256), 256, 0, stream>>>(out_sup, ei_sup, ES,  acc_sup, cnt_sup);
        scatter_add<<<cdiv((long long)ESH * 32, 256), 256, 0, stream>>>(hw,      ei_shp, ESH, acc_shp, cnt_shp);
        scatter_add<<<cdiv((long long)EST * 32, 256), 256, 0, stream>>>(hp,      ei_sto, EST, acc_sto, cnt_sto);

        // product: relu( mean_sup@Wl_sup + bl_sup + mean_shp@Wl_shp + bl_shp
        //               + hp@Wr_sup + hp@Wr_shp )   -- one fused launch, C written once
        gemm_fused<<<cdiv(NP / 16, 8), 256, 0, stream>>>(
            acc_sup, L[layer][2].Wl, cnt_sup,
            acc_shp, L[layer][0].Wl, cnt_shp,
            hp,      L[layer][2].Wr,
            hp,      L[layer][0].Wr,
            L[layer][2].bl, L[layer][0].bl,
            hp_new, NP, 4);

        // warehouse: relu( mean_sto@Wl_sto + bl_sto + hw@Wr_sto )
        gemm_fused<<<cdiv(NW / 16, 8), 256, 0, stream>>>(
            acc_sto, L[layer][1].Wl, cnt_sto,
            hw,      L[layer][1].Wr, nullptr,
            nullptr, nullptr,
            nullptr, nullptr,
            L[layer][1].bl, nullptr,
            hw_new, NW, 2);

        hp = hp_new;
        hw = hw_new;
    }
}